// GraphClassifier_50483045597477
// MI455X (gfx1250) — compile-verified
//
#include <hip/hip_runtime.h>
#include <hip/hip_bf16.h>
#include <cstdint>

#define NNODES 50000
#define NEDGES 800000
#define F_IN   32
#define HID    128
#define TPTS   8
#define KCOEF  101      // DEG+1
#define DEG    100
#define NGRAPH 64
#define NOUT   10
#define CH     512
#define NCHUNK ((NNODES + CH - 1) / CH)   // 98

typedef __attribute__((ext_vector_type(2))) float v2f;
typedef __attribute__((ext_vector_type(8))) float v8f;

// ---------------------------------------------------------------- utilities
__global__ void k_zero_i32(int* __restrict__ p, int n) {
    int i = blockIdx.x * blockDim.x + threadIdx.x;
    if (i < n) p[i] = 0;
}
__global__ void k_zero_f32(float* __restrict__ p, int n) {
    int i = blockIdx.x * blockDim.x + threadIdx.x;
    if (i < n) p[i] = 0.0f;
}

// ---------------------------------------------------------------- CSR build
__global__ void k_count(const int* __restrict__ dst, int* __restrict__ deg) {
    int e = blockIdx.x * blockDim.x + threadIdx.x;
    if (e < NEDGES) atomicAdd(&deg[dst[e]], 1);
}
__global__ void k_partial(const int* __restrict__ deg, int* __restrict__ partial) {
    int c = blockIdx.x * blockDim.x + threadIdx.x;
    if (c < NCHUNK) {
        int b = c * CH;
        int e = b + CH; if (e > NNODES) e = NNODES;
        int s = 0;
        for (int i = b; i < e; ++i) s += deg[i];
        partial[c] = s;
    }
}
__global__ void k_scan_partial(int* __restrict__ partial, int* __restrict__ rowptr) {
    if (blockIdx.x == 0 && threadIdx.x == 0) {
        int run = 0;
        for (int c = 0; c < NCHUNK; ++c) { int v = partial[c]; partial[c] = run; run += v; }
        rowptr[NNODES] = run;
    }
}
__global__ void k_rowptr(const int* __restrict__ deg, const int* __restrict__ partial,
                         int* __restrict__ rowptr) {
    int c = blockIdx.x * blockDim.x + threadIdx.x;
    if (c < NCHUNK) {
        int run = partial[c];
        int b = c * CH;
        int e = b + CH; if (e > NNODES) e = NNODES;
        for (int i = b; i < e; ++i) { rowptr[i] = run; run += deg[i]; }
    }
}
__global__ void k_scatter(const int* __restrict__ src, const int* __restrict__ dst,
                          const float* __restrict__ w, const int* __restrict__ rowptr,
                          int* __restrict__ cursor, int* __restrict__ csrc,
                          float* __restrict__ cw) {
    int e = blockIdx.x * blockDim.x + threadIdx.x;
    if (e < NEDGES) {
        int d = dst[e];
        int p = rowptr[d] + atomicAdd(&cursor[d], 1);
        csrc[p] = src[e];
        cw[p]   = w[e];
    }
}

// ------------------------------------------------- Chebyshev accumulator init
template<int F>
__global__ void k_acc_init(const float* __restrict__ x, const float* __restrict__ coefs,
                           float* __restrict__ acc) {
    int i = blockIdx.x * blockDim.x + threadIdx.x;
    if (i < NNODES * F) {
        int n = i / F, f = i % F;
        float xv = x[i];
        size_t base = (size_t)n * (TPTS * F) + f;
        #pragma unroll
        for (int t = 0; t < TPTS; ++t)
            acc[base + t * F] = coefs[t * KCOEF] * xv;
    }
}

// --------------------------------------------- fused SpMV Chebyshev recursion
// One wave32 per node. VPL = floats per lane (F = 32*VPL).
// Tk = 2 * (sum_e w_e * Tp[src_e]) - Tpp      (FIRST: Tk = sum, the k=1 step)
// napply: how many trailing coefficients to fold into acc at this step
//   (>=1 applies c[k]*Tk; >=2 also c[k-1]*Tp[self]; >=3 also c[k-2]*Tpp[self]).
template<int VPL, bool FIRST>
__global__ void k_cheby_step(const int* __restrict__ rowptr, const int* __restrict__ csrc,
                             const float* __restrict__ cw, const float* __restrict__ Tp,
                             const float* __restrict__ Tpp, float* __restrict__ Tk,
                             float* __restrict__ acc, const float* __restrict__ coefs,
                             int k, int napply) {
    const int F = 32 * VPL;
    int node = (int)((blockIdx.x * blockDim.x + threadIdx.x) >> 5);
    int lane = threadIdx.x & 31;
    if (node >= NNODES) return;

    int beg = rowptr[node], end = rowptr[node + 1];
    float s0 = 0.f, s1 = 0.f, s2 = 0.f, s3 = 0.f;
    const int fo = lane * VPL;

    for (int e0 = beg; e0 < end; e0 += 32) {
        int cnt = end - e0; if (cnt > 32) cnt = 32;
        int   myi = 0;
        float myw = 0.f;
        if (e0 + lane < end) { myi = csrc[e0 + lane]; myw = cw[e0 + lane]; }
        for (int j = 0; j < cnt; ++j) {
            int   sidx = __shfl(myi, j, 32);
            float wj   = __shfl(myw, j, 32);
            const float* vp = Tp + (size_t)sidx * F + fo;
            if (VPL == 4) {
                float4 v = *(const float4*)vp;
                s0 += wj * v.x; s1 += wj * v.y; s2 += wj * v.z; s3 += wj * v.w;
            } else {
                s0 += wj * vp[0];
            }
        }
    }

    size_t base = (size_t)node * F + fo;
    float q0 = 0.f, q1 = 0.f, q2 = 0.f, q3 = 0.f;   // T_{k-2}[self]
    float t0, t1, t2, t3;
    if (FIRST) {
        t0 = s0; t1 = s1; t2 = s2; t3 = s3;
    } else {
        if (VPL == 4) {
            float4 pp = *(const float4*)(Tpp + base);
            q0 = pp.x; q1 = pp.y; q2 = pp.z; q3 = pp.w;
            t0 = 2.f * s0 - q0; t1 = 2.f * s1 - q1;
            t2 = 2.f * s2 - q2; t3 = 2.f * s3 - q3;
        } else {
            q0 = Tpp[base];
            t0 = 2.f * s0 - q0; t1 = t2 = t3 = 0.f;
        }
    }
    if (VPL == 4) *(float4*)(Tk + base) = make_float4(t0, t1, t2, t3);
    else          Tk[base] = t0;

    if (napply > 0) {
        float p0 = 0.f, p1 = 0.f, p2 = 0.f, p3 = 0.f;   // T_{k-1}[self]
        if (napply >= 2) {
            if (VPL == 4) {
                float4 pv = *(const float4*)(Tp + base);
                p0 = pv.x; p1 = pv.y; p2 = pv.z; p3 = pv.w;
            } else {
                p0 = Tp[base];
            }
        }
        size_t abase = (size_t)node * (TPTS * F) + fo;
        #pragma unroll
        for (int t = 0; t < TPTS; ++t) {
            float ck  = coefs[t * KCOEF + k];
            float ck1 = (napply >= 2) ? coefs[t * KCOEF + k - 1] : 0.f;
            float ck2 = (napply >= 3) ? coefs[t * KCOEF + k - 2] : 0.f;
            if (VPL == 4) {
                float4 a = *(const float4*)(acc + abase + t * F);
                a.x += ck * t0 + ck1 * p0 + ck2 * q0;
                a.y += ck * t1 + ck1 * p1 + ck2 * q1;
                a.z += ck * t2 + ck1 * p2 + ck2 * q2;
                a.w += ck * t3 + ck1 * p3 + ck2 * q3;
                *(float4*)(acc + abase + t * F) = a;
            } else {
                acc[abase + t * F] += ck * t0 + ck1 * p0 + ck2 * q0;
            }
        }
    }
}

// ------------------------------------------------- WMMA fp32 GEMM (N == 128)
// C(M,128) = op(A(M,K)) * W(128,K)^T + bias ; one wave per 16x16 tile,
// 8 waves per block cover the 128 output columns. Uses V_WMMA_F32_16X16X4_F32.
template<bool RELU_IN, bool RELU_OUT>
__global__ __launch_bounds__(256)
void k_wmma_gemm128(const float* __restrict__ A, const float* __restrict__ W,
                    const float* __restrict__ bias, float* __restrict__ C, int K) {
    int mbase = blockIdx.x * 16;
    int wv    = threadIdx.x >> 5;      // 0..7 -> column tile
    int lane  = threadIdx.x & 31;
    int l15   = lane & 15;
    int kh    = lane >> 4;             // K sub-pair select (K 0,1 vs 2,3)
    int nbase = wv * 16;

    const float* arow = A + (size_t)(mbase + l15) * K + 2 * kh;
    const float* brow = W + (size_t)(nbase + l15) * K + 2 * kh;
    float bv = bias[nbase + l15];
    v8f c = { bv, bv, bv, bv, bv, bv, bv, bv };

    for (int k0 = 0; k0 < K; k0 += 4) {
        v2f a = *(const v2f*)(arow + k0);
        v2f b = *(const v2f*)(brow + k0);
        if (RELU_IN) { a[0] = fmaxf(a[0], 0.f); a[1] = fmaxf(a[1], 0.f); }
        c = __builtin_amdgcn_wmma_f32_16x16x4_f32(false, a, false, b,
                                                  (short)0, c, false, false);
    }
    #pragma unroll
    for (int r = 0; r < 8; ++r) {
        float v = c[r];
        if (RELU_OUT) v = fmaxf(v, 0.f);
        C[(size_t)(mbase + r + 8 * kh) * HID + nbase + l15] = v;
    }
}

// ------------------------------------------------------------------- pooling
__global__ void k_pool_sum(const float* __restrict__ h, const int* __restrict__ batch,
                           float* __restrict__ sums) {
    int i = blockIdx.x * blockDim.x + threadIdx.x;
    if (i < NNODES * HID) {
        int n = i / HID, f = i % HID;
        atomicAdd(&sums[batch[n] * HID + f], h[i]);
    }
}
__global__ void k_pool_cnt(const int* __restrict__ batch, float* __restrict__ cnt) {
    int n = blockIdx.x * blockDim.x + threadIdx.x;
    if (n < NNODES) atomicAdd(&cnt[batch[n]], 1.0f);
}
__global__ void k_pool_fin(const float* __restrict__ sums, const float* __restrict__ cnt,
                           float* __restrict__ pooled) {
    int i = blockIdx.x * blockDim.x + threadIdx.x;
    if (i < NGRAPH * HID) {
        float c = cnt[i / HID];
        pooled[i] = fmaxf(sums[i] / fmaxf(c, 1.0f), 0.0f);
    }
}

// ---------------------------------------------------------------- classifier
__global__ void k_fc2(const float* __restrict__ z1, const float* __restrict__ Wc2,
                      const float* __restrict__ bc2, float* __restrict__ out) {
    int i = blockIdx.x * blockDim.x + threadIdx.x;
    if (i < NGRAPH * NOUT) {
        int g = i / NOUT, o = i % NOUT;
        float s = bc2[o];
        const float* zr = z1 + g * HID;
        const float* wr = Wc2 + o * HID;
        for (int k = 0; k < HID; ++k) s += zr[k] * wr[k];
        out[i] = s;
    }
}
__global__ void k_logsoftmax(float* __restrict__ out) {
    int g = blockIdx.x * blockDim.x + threadIdx.x;
    if (g < NGRAPH) {
        float* row = out + g * NOUT;
        float m = row[0];
        for (int o = 1; o < NOUT; ++o) m = fmaxf(m, row[o]);
        float s = 0.f;
        for (int o = 0; o < NOUT; ++o) s += __expf(row[o] - m);
        float l = __logf(s) + m;
        for (int o = 0; o < NOUT; ++o) row[o] = row[o] - l;
    }
}

// =========================================================== host launcher
extern "C" void kernel_launch(void* const* d_in, const int* in_sizes, int n_in,
                              void* d_out, int out_size, void* d_ws, size_t ws_size,
                              hipStream_t stream) {
    (void)in_sizes; (void)n_in; (void)out_size; (void)ws_size;
    const float* x     = (const float*)d_in[0];
    const int*   eidx  = (const int*)d_in[1];
    const float* ew    = (const float*)d_in[2];
    const int*   batch = (const int*)d_in[3];
    const float* coefs = (const float*)d_in[4];
    const float* W1    = (const float*)d_in[5];
    const float* b1    = (const float*)d_in[6];
    const float* W2    = (const float*)d_in[7];
    const float* b2    = (const float*)d_in[8];
    const float* Wc1   = (const float*)d_in[9];
    const float* bc1   = (const float*)d_in[10];
    const float* Wc2   = (const float*)d_in[11];
    const float* bc2   = (const float*)d_in[12];
    const int* esrc = eidx;
    const int* edst = eidx + NEDGES;

    char* wsb = (char*)d_ws;
    size_t off = 0;
    auto carve = [&](size_t bytes) -> char* {
        char* p = wsb + off;
        off = (off + bytes + 255) & ~(size_t)255;
        return p;
    };
    int*   deg     = (int*)  carve((size_t)NNODES * 4);
    int*   cursor  = (int*)  carve((size_t)NNODES * 4);
    int*   partial = (int*)  carve((size_t)NCHUNK * 4);
    int*   rowptr  = (int*)  carve((size_t)(NNODES + 1) * 4);
    int*   csrc    = (int*)  carve((size_t)NEDGES * 4);
    float* cw      = (float*)carve((size_t)NEDGES * 4);
    float* bufA    = (float*)carve((size_t)NNODES * HID * 4);
    float* bufB    = (float*)carve((size_t)NNODES * HID * 4);
    float* bufC    = (float*)carve((size_t)NNODES * HID * 4);
    float* h1      = (float*)carve((size_t)NNODES * HID * 4);
    float* acc2    = (float*)carve((size_t)NNODES * TPTS * HID * 4);
    float* acc1    = acc2;   // aliased: acc1 (N x 256) lives inside acc2 region
    float* sums    = (float*)carve((size_t)NGRAPH * HID * 4);
    float* cnt     = (float*)carve((size_t)NGRAPH * 4);
    float* pooled  = (float*)carve((size_t)NGRAPH * HID * 4);
    float* z1      = (float*)carve((size_t)NGRAPH * HID * 4);
    float* h2      = bufA;   // free after second Chebyshev loop
    float* outf    = (float*)d_out;

    // ---- build CSR (by destination) -------------------------------------
    k_zero_i32<<<(NNODES + 255) / 256, 256, 0, stream>>>(deg, NNODES);
    k_zero_i32<<<(NNODES + 255) / 256, 256, 0, stream>>>(cursor, NNODES);
    k_count<<<(NEDGES + 255) / 256, 256, 0, stream>>>(edst, deg);
    k_partial<<<1, 128, 0, stream>>>(deg, partial);
    k_scan_partial<<<1, 1, 0, stream>>>(partial, rowptr);
    k_rowptr<<<1, 128, 0, stream>>>(deg, partial, rowptr);
    k_scatter<<<(NEDGES + 255) / 256, 256, 0, stream>>>(esrc, edst, ew, rowptr,
                                                        cursor, csrc, cw);

    const int gridw = (NNODES + 7) / 8;   // 8 waves per 256-thread block
    float* ring[3] = { bufA, bufB, bufC };
    auto nap = [](int k) { return (k == DEG) ? 1 : ((k % 3 == 0) ? 3 : 0); };

    // ---- Chebyshev pass 1 (F = 32) --------------------------------------
    k_acc_init<F_IN><<<(NNODES * F_IN + 255) / 256, 256, 0, stream>>>(x, coefs, acc1);
    k_cheby_step<1, true><<<gridw, 256, 0, stream>>>(rowptr, csrc, cw, x, x,
                                                     ring[0], acc1, coefs, 1, nap(1));
    for (int k = 2; k <= DEG; ++k) {
        const float* Tpp = (k == 2) ? x : (const float*)ring[(k - 3) % 3];
        k_cheby_step<1, false><<<gridw, 256, 0, stream>>>(rowptr, csrc, cw,
            ring[(k - 2) % 3], Tpp, ring[(k - 1) % 3], acc1, coefs, k, nap(k));
    }

    // ---- GEMM1: relu(relu(acc1) @ W1^T + b1) -> h1 ----------------------
    k_wmma_gemm128<true, true><<<NNODES / 16, 256, 0, stream>>>(acc1, W1, b1, h1,
                                                                TPTS * F_IN);

    // ---- Chebyshev pass 2 (F = 128) -------------------------------------
    k_acc_init<HID><<<(NNODES * HID + 255) / 256, 256, 0, stream>>>(h1, coefs, acc2);
    k_cheby_step<4, true><<<gridw, 256, 0, stream>>>(rowptr, csrc, cw, h1, h1,
                                                     ring[0], acc2, coefs, 1, nap(1));
    for (int k = 2; k <= DEG; ++k) {
        const float* Tpp = (k == 2) ? h1 : (const float*)ring[(k - 3) % 3];
        k_cheby_step<4, false><<<gridw, 256, 0, stream>>>(rowptr, csrc, cw,
            ring[(k - 2) % 3], Tpp, ring[(k - 1) % 3], acc2, coefs, k, nap(k));
    }

    // ---- GEMM2: relu(acc2) @ W2^T + b2 -> h2 (no output relu) -----------
    k_wmma_gemm128<true, false><<<NNODES / 16, 256, 0, stream>>>(acc2, W2, b2, h2,
                                                                 TPTS * HID);

    // ---- mean pooling per graph -----------------------------------------
    k_zero_f32<<<(NGRAPH * HID + 255) / 256, 256, 0, stream>>>(sums, NGRAPH * HID);
    k_zero_f32<<<1, NGRAPH, 0, stream>>>(cnt, NGRAPH);
    k_pool_sum<<<(NNODES * HID + 255) / 256, 256, 0, stream>>>(h2, batch, sums);
    k_pool_cnt<<<(NNODES + 255) / 256, 256, 0, stream>>>(batch, cnt);
    k_pool_fin<<<(NGRAPH * HID + 255) / 256, 256, 0, stream>>>(sums, cnt, pooled);

    // ---- classifier ------------------------------------------------------
    k_wmma_gemm128<false, true><<<NGRAPH / 16, 256, 0, stream>>>(pooled, Wc1, bc1,
                                                                 z1, HID);
    k_fc2<<<(NGRAPH * NOUT + 255) / 256, 256, 0, stream>>>(z1, Wc2, bc2, outf);
    k_logsoftmax<<<1, NGRAPH, 0, stream>>>(outf);
}